// Modulated_FC_Conv_34445637714446
// MI455X (gfx1250) — compile-verified
//
#include <hip/hip_runtime.h>

// Problem constants (match reference)
#define BDIM 8
#define NPTS 131072
#define INC  64
#define OUTC 64
#define SDIM 256

typedef __attribute__((ext_vector_type(2))) float v2f;
typedef __attribute__((ext_vector_type(8))) float v8f;
typedef int v4i_t __attribute__((vector_size(16)));

// Weight region layout (floats), per batch:
//   half0: even K-pairs  -> [16 rows][64 cols][2]   = 2048 floats
//   pad  : 32 floats (128 B -> shifts half1 by 32 LDS banks)
//   half1: odd  K-pairs  -> [16 rows][64 cols][2]   = 2048 floats
#define WREG 4128                    // floats per batch region
#define WHALF 2080                   // float offset of half1 (2048 + 32)

#if defined(__has_builtin)
#if __has_builtin(__builtin_amdgcn_global_load_async_to_lds_b128)
#define HAVE_ASYNC_LDS 1
#endif
#if __has_builtin(__builtin_amdgcn_s_wait_asynccnt)
#define HAVE_WAIT_ASYNC 1
#endif
#endif

#if defined(HAVE_ASYNC_LDS)
// (as1 int4* gsrc, as3 int4* ldst, imm offset, imm cpol) per clang diagnostic
#define ASYNC_LDS_B128(gsrc, ldst)                                        \
  __builtin_amdgcn_global_load_async_to_lds_b128(                         \
      (__attribute__((address_space(1))) v4i_t*)(gsrc),                   \
      (__attribute__((address_space(3))) v4i_t*)(ldst), 0, 0)
#endif

__device__ __forceinline__ void async_drain() {
#if defined(HAVE_ASYNC_LDS)
#if defined(HAVE_WAIT_ASYNC)
  __builtin_amdgcn_s_wait_asynccnt(0);
#else
  asm volatile("s_wait_asynccnt 0" ::: "memory");
#endif
#endif
}

// ---------------------------------------------------------------------------
// Kernel 1: modulation + demodulation -> per-batch weights, written directly
// in the K-pair-interleaved WMMA B layout described above.
// ---------------------------------------------------------------------------
__global__ __launch_bounds__(64)
void modfc_prep(const float* __restrict__ style,
                const float* __restrict__ weight,      // (out_c, in_c)
                const float* __restrict__ mod_weight,  // (in_c, style_dim)
                const float* __restrict__ mod_bias,    // (in_c)
                float* __restrict__ wP) {              // (B, WREG)
  __shared__ float s[INC];
  const int b   = blockIdx.x;
  const int tid = threadIdx.x;  // 0..63

  // s[i] = style . modW[i] * (1/sqrt(256)) + mod_bias[i] + 1
  {
    const float* st = style + b * SDIM;
    const float* mw = mod_weight + tid * SDIM;
    float acc = 0.f;
#pragma unroll 8
    for (int d = 0; d < SDIM; ++d) acc += st[d] * mw[d];
    s[tid] = acc * 0.0625f + mod_bias[tid] + 1.0f;
  }
  __syncthreads();

  const int o = tid;
  float dacc = 0.f;
#pragma unroll
  for (int i = 0; i < INC; ++i) {
    float wv = 0.125f * weight[o * INC + i] * s[i];   // 1/sqrt(64) = 0.125
    dacc += wv * wv;
  }
  const float dem = rsqrtf(dacc + 1e-8f);

  float* reg = wP + (size_t)b * WREG;
#pragma unroll
  for (int i = 0; i < INC; ++i) {
    // pair p = i/2, element e = i%2; half = p&1, row = p>>1
    const int idx = ((i >> 1) & 1) * WHALF + (i >> 2) * 128 + o * 2 + (i & 1);
    reg[idx] = 0.125f * weight[o * INC + i] * s[i] * dem;
  }
}

// ---------------------------------------------------------------------------
// Kernel 2: out[b] = x[b] (N x 64) @ w[b].T + bias via fp32 WMMA 16x16x4.
// 256 threads = 8 wave32; block tile = 128 rows; wave tile = 16 rows x 64 out.
// ---------------------------------------------------------------------------
#define ROWS 128
#define XS_S 68   // padded x-tile row stride (floats)

__global__ __launch_bounds__(256)
void modfc_gemm(const float* __restrict__ x,      // (B, N, in_c)
                const float* __restrict__ bias,   // (out_c)
                const float* __restrict__ wP,     // (B, WREG)
                float* __restrict__ out) {        // (B, N, out_c)
  __shared__ float xs[ROWS * XS_S];   // 34816 B
  __shared__ float wsh[WREG];         // 16512 B

  const int b       = blockIdx.y;
  const int rowBase = blockIdx.x * ROWS;
  const int t       = threadIdx.x;

  // --- stage weight region (flat copy, pads included) ---
  {
    const float4* wg = (const float4*)(wP + (size_t)b * WREG);
    for (int f4 = t; f4 < WREG / 4; f4 += 256) {      // 1032 float4s
#if defined(HAVE_ASYNC_LDS)
      ASYNC_LDS_B128(wg + f4, &wsh[f4 * 4]);
#else
      *(float4*)&wsh[f4 * 4] = wg[f4];
#endif
    }
  }
  // --- stage x tile (128 rows x 64 K), coalesced 16B per lane ---
  {
    const float4* xg = (const float4*)(x + ((size_t)b * NPTS + rowBase) * INC);
#pragma unroll
    for (int p = 0; p < 8; ++p) {
      int f4 = p * 256 + t;            // 0..2047 float4s (linear in global)
      int r  = f4 >> 4;                // local row
      int k4 = (f4 & 15) * 4;          // K group
#if defined(HAVE_ASYNC_LDS)
      ASYNC_LDS_B128(xg + f4, &xs[r * XS_S + k4]);
#else
      *(float4*)&xs[r * XS_S + k4] = xg[f4];
#endif
    }
  }
  async_drain();
  __syncthreads();

  const int wave = t >> 5;
  const int lane = t & 31;
  const int rL   = lane & 15;
  const bool lo  = (lane < 16);
  const int c0   = lo ? 0 : 2;         // K sub-offset for this lane group

  // C init = bias broadcast along rows
  v8f acc[4];
#pragma unroll
  for (int t0 = 0; t0 < 4; ++t0) {
    float bv = bias[t0 * 16 + rL];
    v8f c = {bv, bv, bv, bv, bv, bv, bv, bv};
    acc[t0] = c;
  }

  // A fragment base: lane holds row (wave*16 + rL), K pair starting at c0
  const float* aPtr = &xs[(wave * 16 + rL) * XS_S + c0];
  // B fragment base: half = c0/2 selects even/odd K-pair region
  const v2f* bPtr = (const v2f*)&wsh[(lo ? 0 : WHALF) + rL * 2];

#pragma unroll
  for (int j = 0; j < 16; ++j) {            // K steps of 4
    v2f a = *(const v2f*)(aPtr + 4 * j);    // K = 4j+c0, 4j+c0+1
#pragma unroll
    for (int t0 = 0; t0 < 4; ++t0) {        // 4 output tiles of 16 cols
      v2f bf = bPtr[j * 64 + t0 * 16];      // (K pair 2j+half, col 16*t0+rL)
      acc[t0] = __builtin_amdgcn_wmma_f32_16x16x4_f32(
          /*neg_a=*/false, a, /*neg_b=*/false, bf,
          /*c_mod=*/(short)0, acc[t0],
          /*reuse_a=*/false, /*reuse_b=*/false);
    }
  }

  // --- store D: VGPR v -> row v (lanes 0-15) / v+8 (lanes 16-31), col = lane
  const int rowTop = rowBase + wave * 16 + (lo ? 0 : 8);
  float* orow = out + ((size_t)b * NPTS + rowTop) * OUTC;
#pragma unroll
  for (int t0 = 0; t0 < 4; ++t0) {
    const int col = t0 * 16 + rL;
#pragma unroll
    for (int v = 0; v < 8; ++v) {
      orow[(size_t)v * OUTC + col] = acc[t0][v];
    }
  }
}

// ---------------------------------------------------------------------------
extern "C" void kernel_launch(void* const* d_in, const int* in_sizes, int n_in,
                              void* d_out, int out_size, void* d_ws, size_t ws_size,
                              hipStream_t stream) {
  const float* x          = (const float*)d_in[0];
  const float* style      = (const float*)d_in[1];
  const float* weight     = (const float*)d_in[2];
  const float* bias       = (const float*)d_in[3];
  const float* mod_weight = (const float*)d_in[4];
  const float* mod_bias   = (const float*)d_in[5];
  float* out = (float*)d_out;
  float* wP  = (float*)d_ws;   // B * WREG floats = ~132 KB scratch

  modfc_prep<<<dim3(BDIM), dim3(64), 0, stream>>>(style, weight, mod_weight,
                                                  mod_bias, wP);
  modfc_gemm<<<dim3(NPTS / ROWS, BDIM), dim3(256), 0, stream>>>(x, bias, wP,
                                                                out);
}